// Q_real_M_9311489098219
// MI455X (gfx1250) — compile-verified
//
#include <hip/hip_runtime.h>

#define NPTS   262144          // number of veronese columns == number of outputs
#define NMASK  (NPTS - 1)      // NPTS = 2^18
#define DIMS   10
#define DVER   286             // veronese length = C(13,3)
#define DPAD   288             // padded to multiple of 16 (and of 4 for K)
#define KCHUNKS 72             // 288 / 4
#define CTILES  18             // 288 / 16
#define ROWS    64             // output rows per block
#define TPB     128            // 4 waves (wave32)
#define LDS_STRIDE 292         // 292 % 64 = 36 -> conflict-free ds_load_b64 column reads

typedef float v2f __attribute__((ext_vector_type(2)));
typedef float v8f __attribute__((ext_vector_type(8)));

// ---------------------------------------------------------------------------
// Kernel 0: zero-pad M_inv (286x286) into Mpad (288x288) in workspace.
// ---------------------------------------------------------------------------
__global__ void pad_m_kernel(const float* __restrict__ M, float* __restrict__ Mpad) {
    int i = blockIdx.x * blockDim.x + threadIdx.x;
    if (i >= DPAD * DPAD) return;
    int r = i / DPAD;
    int c = i - r * DPAD;
    Mpad[i] = (r < DVER && c < DVER) ? M[r * DVER + c] : 0.0f;
}

// ---------------------------------------------------------------------------
// Main kernel: fused veronese-tile construction + WMMA quadratic form.
// Each block: 64 outputs. Each wave: one 16-row tile, full 288-col sweep.
// ---------------------------------------------------------------------------
__global__ void __launch_bounds__(TPB) quadform_kernel(const float* __restrict__ x,
                                                       const float* __restrict__ Mpad,
                                                       float* __restrict__ out) {
    extern __shared__ float smem[];
    float*    Wl   = smem;                                   // ROWS x LDS_STRIDE
    unsigned* mono = (unsigned*)(smem + ROWS * LDS_STRIDE);  // 286 packed entries

    const int tid = threadIdx.x;
    const int n0  = blockIdx.x * ROWS;

    // ---- build monomial index table: packed = deg<<12 | i0 | i1<<4 | i2<<8 ----
    for (int r = tid; r < DVER; r += TPB) {
        unsigned p = 0u;                           // r == 0 : constant 1
        if (r >= 1 && r <= 10) {
            p = (1u << 12) | (unsigned)(r - 1);
        } else if (r >= 11 && r <= 65) {           // x_j * x_k, j <= k
            int q = r - 11, j = 0;
            while (q >= 10 - j) { q -= 10 - j; ++j; }
            p = (2u << 12) | (unsigned)j | ((unsigned)(j + q) << 4);
        } else if (r >= 66) {                      // x_j * x_k * x_l, j <= k <= l
            int q = r - 66, j = 0;
            while (q >= (10 - j) * (11 - j) / 2) { q -= (10 - j) * (11 - j) / 2; ++j; }
            int k = j;
            while (q >= 10 - k) { q -= 10 - k; ++k; }
            p = (3u << 12) | (unsigned)j | ((unsigned)k << 4) | ((unsigned)(k + q) << 8);
        }
        mono[r] = p;
    }
    __syncthreads();

    // ---- fill W tile: Wl[row][d] = v_flat[(n0+row)*286 + d], zero-padded ----
    // v_flat[g]: monomial row g>>18 at column g & NMASK.
    for (int e = tid; e < ROWS * DPAD; e += TPB) {           // 64*288/128 = 144 iters, uniform
        int row = e / DPAD;
        int d   = e - row * DPAD;
        float val = 0.0f;
        if (d < DVER) {
            unsigned g = (unsigned)(n0 + row) * DVER + (unsigned)d;
            int r = (int)(g >> 18);
            int c = (int)(g & NMASK);
            unsigned p = mono[r];
            unsigned deg = p >> 12;
            val = 1.0f;
            if (deg > 0u) val  = x[(p & 15u) * NPTS + c];
            if (deg > 1u) val *= x[((p >> 4) & 15u) * NPTS + c];
            if (deg > 2u) val *= x[((p >> 8) & 15u) * NPTS + c];
        }
        Wl[row * LDS_STRIDE + d] = val;
    }
    __syncthreads();

    // ---- quadratic form via WMMA: T = W @ Mpad, out[n] = sum_d T[n,d]*W[n,d] ----
    const int wave  = tid >> 5;        // 0..3, owns rows [wave*16, wave*16+16)
    const int lane  = tid & 31;
    const int lhalf = lane >> 4;       // 0: lanes 0-15, 1: lanes 16-31
    const int l16   = lane & 15;
    const int row0  = wave * 16;

    // A fragment base: lane m (0-15) -> row row0+m, K-pair k0+{0,1};
    //                  lane m+16    -> row row0+m, K-pair k0+{2,3}.  (16x4 f32 A layout)
    const float* a_base = &Wl[(row0 + l16) * LDS_STRIDE + lhalf * 2];
    // B fragment base: lanes 0-15 -> rows k0+{0,1}, lanes 16-31 -> rows k0+{2,3}; col = l16.
    const float* b_col  = &Mpad[(lhalf * 2) * DPAD];

    v8f acc = {0.f, 0.f, 0.f, 0.f, 0.f, 0.f, 0.f, 0.f};

    for (int ct = 0; ct < CTILES; ++ct) {
        const int col0 = ct * 16;
        const float* b_base = b_col + col0 + l16;

        v8f c = {0.f, 0.f, 0.f, 0.f, 0.f, 0.f, 0.f, 0.f};
#pragma unroll 4
        for (int kc = 0; kc < KCHUNKS; ++kc) {
            const int k0 = kc * 4;
            v2f a = *(const v2f*)(a_base + k0);                       // ds_load_b64
            v2f b = { b_base[k0 * DPAD], b_base[k0 * DPAD + DPAD] };  // L2-resident M
            c = __builtin_amdgcn_wmma_f32_16x16x4_f32(
                    /*neg_a=*/false, a, /*neg_b=*/false, b,
                    /*c_mod=*/(short)0, c, /*reuse_a=*/false, /*reuse_b=*/false);
        }

        // C/D layout: VGPR v, lanes 0-15 -> (row0+v, col0+l16); lanes 16-31 -> (row0+v+8, ...)
#pragma unroll
        for (int v = 0; v < 8; ++v) {
            int row = row0 + v + lhalf * 8;
            acc[v] += c[v] * Wl[row * LDS_STRIDE + col0 + l16];
        }
    }

    // ---- reduce over the 16 columns held per 16-lane half, write out ----
#pragma unroll
    for (int v = 0; v < 8; ++v) {
        float s = acc[v];
        s += __shfl_xor(s, 1, 32);
        s += __shfl_xor(s, 2, 32);
        s += __shfl_xor(s, 4, 32);
        s += __shfl_xor(s, 8, 32);
        if (l16 == 0) out[n0 + row0 + v + lhalf * 8] = s;
    }
}

// ---------------------------------------------------------------------------
extern "C" void kernel_launch(void* const* d_in, const int* in_sizes, int n_in,
                              void* d_out, int out_size, void* d_ws, size_t ws_size,
                              hipStream_t stream) {
    (void)in_sizes; (void)n_in; (void)out_size; (void)ws_size;
    const float* x = (const float*)d_in[0];     // (262144, 10) f32 (flat)
    const float* M = (const float*)d_in[1];     // (286, 286) f32
    float* out  = (float*)d_out;                // 262144 f32
    float* Mpad = (float*)d_ws;                 // 288*288 f32 = 331,776 B

    pad_m_kernel<<<(DPAD * DPAD + 255) / 256, 256, 0, stream>>>(M, Mpad);

    const size_t lds_bytes = (size_t)(ROWS * LDS_STRIDE + DVER) * sizeof(float);
    quadform_kernel<<<NPTS / ROWS, TPB, lds_bytes, stream>>>(x, Mpad, out);
}